// Attention_78314433675979
// MI455X (gfx1250) — compile-verified
//
#include <hip/hip_runtime.h>
#include <hip/hip_bf16.h>

// Problem constants (from reference)
#define T_    600
#define B_    128
#define C_    512
#define QH_   1024
#define H_    128
#define K_    9
#define PAD_  4
#define WL_   129           // window length
#define FULL_ (T_ + 2*PAD_) // 608
#define AVG_  1.5f

typedef __attribute__((ext_vector_type(2))) float v2f;
typedef __attribute__((ext_vector_type(8))) float v8f;

// ---------------------------------------------------------------------------
// Kernel 1: q = query[0] @ Wq^T + bq   (M=B=128, N=H=128, K=QH=1024), fp32 WMMA
// One 16x16 tile per wave, 4 waves/block, 16 blocks = 64 tiles.
// A-frag (16x4 f32): v[0]=K0 (lanes0-15)/K2 (lanes16-31), v[1]=K1/K3.
// B-frag (4x16 f32): same K split, N across lanes. D: VGPR r -> M=r+8*(lane>=16).
// ---------------------------------------------------------------------------
__global__ __launch_bounds__(128)
void qproj_wmma_kernel(const float* __restrict__ Qin,   // [B, QH]
                       const float* __restrict__ Wq,    // [H, QH]
                       const float* __restrict__ bq,    // [H]
                       float* __restrict__ qout)        // [B, H]
{
    const int lane = threadIdx.x & 31;
    const int wave = threadIdx.x >> 5;
    const int tile = blockIdx.x * 4 + wave;     // 0..63
    const int m0   = (tile >> 3) << 4;          // B-tile base
    const int n0   = (tile & 7) << 4;           // H-tile base
    const int half = lane >> 4;                 // 0: lanes 0-15, 1: lanes 16-31
    const int mr   = m0 + (lane & 15);
    const int nr   = n0 + (lane & 15);
    const int koff = half * 2;

    v8f c0 = {}, c1 = {}, c2 = {}, c3 = {};
    const float* aBase = Qin + (size_t)mr * QH_ + koff;
    const float* bBase = Wq  + (size_t)nr * QH_ + koff;

    for (int k0 = 0; k0 < QH_; k0 += 16) {
        v2f a0 = *(const v2f*)(aBase + k0);
        v2f b0 = *(const v2f*)(bBase + k0);
        v2f a1 = *(const v2f*)(aBase + k0 + 4);
        v2f b1 = *(const v2f*)(bBase + k0 + 4);
        v2f a2 = *(const v2f*)(aBase + k0 + 8);
        v2f b2 = *(const v2f*)(bBase + k0 + 8);
        v2f a3 = *(const v2f*)(aBase + k0 + 12);
        v2f b3 = *(const v2f*)(bBase + k0 + 12);
        c0 = __builtin_amdgcn_wmma_f32_16x16x4_f32(false, a0, false, b0, (short)0, c0, false, false);
        c1 = __builtin_amdgcn_wmma_f32_16x16x4_f32(false, a1, false, b1, (short)0, c1, false, false);
        c2 = __builtin_amdgcn_wmma_f32_16x16x4_f32(false, a2, false, b2, (short)0, c2, false, false);
        c3 = __builtin_amdgcn_wmma_f32_16x16x4_f32(false, a3, false, b3, (short)0, c3, false, false);
    }

    const int   n   = n0 + (lane & 15);
    const float bqv = bq[n];
#pragma unroll
    for (int r = 0; r < 8; ++r) {
        const int m = m0 + r + half * 8;
        qout[(size_t)m * H_ + n] = ((c0[r] + c1[r]) + (c2[r] + c3[r])) + bqv;
    }
}

// ---------------------------------------------------------------------------
// Kernel 2: per-batch conv + tanh-score + masked softmax + context + scatter.
// One block (256 threads = 8 wave32) per batch element.
// ---------------------------------------------------------------------------
__global__ __launch_bounds__(256)
void attn_ctx_kernel(const float* __restrict__ tokens,        // [T, B, C]
                     const unsigned char* __restrict__ tmask, // [B, T] (bool)
                     const int* __restrict__ num_tokens,      // [B]
                     const float* __restrict__ cum,           // [B, T+2p]
                     const int* __restrict__ wstart,          // [B]
                     const float* __restrict__ conv_w,        // [H, 1, K]
                     const float* __restrict__ conv_b,        // [H]
                     const float* __restrict__ Wsv,           // [H]
                     const float* __restrict__ qbuf,          // [B, H]
                     float* __restrict__ out_ctx,             // [B, C]
                     float* __restrict__ out_align,           // [B, T]
                     float* __restrict__ out_cum,             // [B, T+2p]
                     float* __restrict__ out_ws)              // [B] (as float)
{
    const int b    = blockIdx.x;
    const int tid  = threadIdx.x;
    const int lane = tid & 31;
    const int wave = tid >> 5;

    __shared__ float xw[WL_ + 2*PAD_];   // conv input window (137)
    __shared__ float qb[H_];             // q[b,h] + conv_b[h]
    __shared__ float cw[H_ * K_];        // conv weights
    __shared__ float wss[H_];            // Ws
    __shared__ float sc[WL_];            // scores -> alignment
    __shared__ float redf[8];
    __shared__ int   redi[8];
    __shared__ float s_max, s_sum;
    __shared__ int   s_argmax;

    const int ws = wstart[b];

    // ---- stage inputs into LDS ----
    if (tid < WL_ + 2*PAD_)
        xw[tid] = cum[(size_t)b * FULL_ + ws + tid] * (1.0f / AVG_) - 1.0f;
    if (tid < H_) {
        qb[tid]  = qbuf[(size_t)b * H_ + tid] + conv_b[tid];
        wss[tid] = Wsv[tid];
    }
    for (int i = tid; i < H_ * K_; i += 256) cw[i] = conv_w[i];
    __syncthreads();

    // ---- scores: sc[l] = sum_h Ws[h]*tanh(conv(l,h) + q[b,h]) , masked ----
    for (int l = tid; l < WL_; l += 256) {
        float s = 0.0f;
        for (int h = 0; h < H_; ++h) {
            float acc = qb[h];
            const float* w = &cw[h * K_];
#pragma unroll
            for (int k = 0; k < K_; ++k) acc = fmaf(w[k], xw[l + k], acc);
            s = fmaf(wss[h], tanhf(acc), s);
        }
        sc[l] = tmask[(size_t)b * T_ + ws + l] ? s : -__builtin_inff();
    }
    __syncthreads();

    // ---- combined max + argmax (first index of max) reduction ----
    float bv = -__builtin_inff();
    int   bi = 0x7fffffff;
    for (int l = tid; l < WL_; l += 256) {
        const float v = sc[l];
        if (v > bv || (v == bv && l < bi)) { bv = v; bi = l; }
    }
    for (int off = 16; off; off >>= 1) {
        const float ov = __shfl_xor(bv, off, 32);
        const int   oi = __shfl_xor(bi, off, 32);
        if (ov > bv || (ov == bv && oi < bi)) { bv = ov; bi = oi; }
    }
    if (lane == 0) { redf[wave] = bv; redi[wave] = bi; }
    __syncthreads();
    if (tid == 0) {
        float tv = redf[0]; int ti = redi[0];
        for (int i = 1; i < 8; ++i)
            if (redf[i] > tv || (redf[i] == tv && redi[i] < ti)) { tv = redf[i]; ti = redi[i]; }
        s_max = tv; s_argmax = ti;
        // window advance: argmax(full) = ws + PAD + ti -> new_ws = ws + ti - WL/2
        int nw = ws + ti - (WL_ / 2);
        const int lim = num_tokens[b] - WL_;
        nw = (nw < lim) ? nw : lim;
        nw = (nw > 0) ? nw : 0;
        nw = (nw > ws) ? nw : ws;
        out_ws[b] = (float)nw;
    }
    __syncthreads();

    // ---- exp + sum ----
    const float mx = s_max;
    float ls = 0.0f;
    for (int l = tid; l < WL_; l += 256) {
        const float e = expf(sc[l] - mx);
        sc[l] = e;
        ls += e;
    }
    for (int off = 16; off; off >>= 1) ls += __shfl_xor(ls, off, 32);
    if (lane == 0) redf[wave] = ls;
    __syncthreads();
    if (tid == 0) {
        float t = 0.0f;
        for (int i = 0; i < 8; ++i) t += redf[i];
        s_sum = t;
    }
    __syncthreads();
    const float inv = 1.0f / s_sum;
    for (int l = tid; l < WL_; l += 256) sc[l] *= inv;
    __syncthreads();

    // ---- scatter alignment into padded layouts: out_align, out_cum ----
    const int ofs = ws + PAD_;
    for (int j = tid; j < FULL_; j += 256) {
        const int l = j - ofs;
        const float f = (l >= 0 && l < WL_) ? sc[l] : 0.0f;
        out_cum[(size_t)b * FULL_ + j] = cum[(size_t)b * FULL_ + j] + f;
        if (j >= PAD_ && j < T_ + PAD_) out_align[(size_t)b * T_ + (j - PAD_)] = f;
    }

    // ---- context: bandwidth-dominant streaming of the token window ----
    // 256 threads x float2 = one full 512-float row per iteration (coalesced).
    {
        const int c2 = tid * 2;
        float ax = 0.0f, ay = 0.0f;
        const float* tb = tokens + (size_t)ws * (B_ * C_) + (size_t)b * C_ + c2;
        const size_t rowStride = (size_t)B_ * C_;
        for (int l = 0; l < WL_; ++l) {
            if (l + 4 < WL_) __builtin_prefetch(tb + (size_t)(l + 4) * rowStride, 0, 0);
            const float a = sc[l];
            const v2f tv = *(const v2f*)(tb + (size_t)l * rowStride);
            ax = fmaf(a, tv.x, ax);
            ay = fmaf(a, tv.y, ay);
        }
        v2f r; r.x = ax; r.y = ay;
        *(v2f*)(out_ctx + (size_t)b * C_ + c2) = r;
    }
}

// ---------------------------------------------------------------------------
// Output layout (flat f32, return order):
//   context  [128,512]  offset 0
//   align    [128,600]  offset 65536
//   new_cum  [128,608]  offset 142336
//   new_ws   [128]      offset 220160
// ---------------------------------------------------------------------------
extern "C" void kernel_launch(void* const* d_in, const int* in_sizes, int n_in,
                              void* d_out, int out_size, void* d_ws, size_t ws_size,
                              hipStream_t stream) {
    (void)in_sizes; (void)n_in; (void)out_size; (void)ws_size;
    const float*         tokens  = (const float*)d_in[0];
    const unsigned char* tmask   = (const unsigned char*)d_in[1];
    const int*           ntok    = (const int*)d_in[2];
    const float*         query   = (const float*)d_in[3];   // [1,B,QH] -> [B,QH]
    const float*         cum     = (const float*)d_in[4];
    const int*           wstart  = (const int*)d_in[5];
    const float*         conv_w  = (const float*)d_in[6];
    const float*         conv_b  = (const float*)d_in[7];
    const float*         Wq      = (const float*)d_in[8];
    const float*         bq      = (const float*)d_in[9];
    const float*         Wsv     = (const float*)d_in[10];

    float* out  = (float*)d_out;
    float* qbuf = (float*)d_ws;                 // [B,H] = 64 KB scratch

    qproj_wmma_kernel<<<16, 128, 0, stream>>>(query, Wq, bq, qbuf);

    float* out_ctx   = out;
    float* out_align = out + (size_t)B_ * C_;                                   // 65536
    float* out_cum   = out + (size_t)B_ * C_ + (size_t)B_ * T_;                 // 142336
    float* out_ws    = out + (size_t)B_ * C_ + (size_t)B_ * T_ + (size_t)B_ * FULL_; // 220160

    attn_ctx_kernel<<<B_, 256, 0, stream>>>(tokens, tmask, ntok, cum, wstart,
                                            conv_w, conv_b, Wsv, qbuf,
                                            out_ctx, out_align, out_cum, out_ws);
}